// AdaptiveLocal2DLayer_59717225284016
// MI455X (gfx1250) — compile-verified
//
#include <hip/hip_runtime.h>

typedef __attribute__((ext_vector_type(2))) float v2f;
typedef __attribute__((ext_vector_type(8))) float v8f;
typedef __attribute__((__vector_size__(4 * sizeof(int)))) int v4i;

#define B_    32
#define C_    3
#define H_    128
#define W_    128
#define N_    1024
#define K_    (C_ * H_ * W_)        // 49152
#define KSLICES 16
#define KSLICE_LEN (K_ / KSLICES)   // 3072
#define KC    64                    // k-chunk staged in LDS (double buffered)
#define NCHUNK (KSLICE_LEN / KC)    // 48
#define NT_PER_WG 32                // neurons per workgroup (2 waves x 16)

#define GLOBAL_AS __attribute__((address_space(1)))
#define LDS_AS    __attribute__((address_space(3)))

// ---- CDNA5 async global->LDS copy (ASYNCcnt-tracked), with safe fallback ----
#if __has_builtin(__builtin_amdgcn_global_load_async_to_lds_b128)
#define HAVE_ASYNC_LDS 1
#define ASYNC_COPY_B128(dst, src)                                              \
  __builtin_amdgcn_global_load_async_to_lds_b128((GLOBAL_AS v4i*)(src),        \
                                                 (LDS_AS v4i*)(dst), 0, 0)
#else
#define HAVE_ASYNC_LDS 0
#define ASYNC_COPY_B128(dst, src) (*(float4*)(dst) = *(const float4*)(src))
#endif

__device__ __forceinline__ void async_wait_all() {
#if HAVE_ASYNC_LDS
#if __has_builtin(__builtin_amdgcn_s_wait_asynccnt)
  __builtin_amdgcn_s_wait_asynccnt(0);
#else
  asm volatile("s_wait_asynccnt 0x0" ::: "memory");
#endif
#endif
}

__global__ void zero_out_kernel(float* __restrict__ out, int n) {
  int i = blockIdx.x * blockDim.x + threadIdx.x;
  if (i < n) out[i] = 0.0f;
}

__global__ __launch_bounds__(64) void adaptive_local2d_wmma_kernel(
    const float* __restrict__ x, const float* __restrict__ mu_x,
    const float* __restrict__ mu_y, const float* __restrict__ sig_x,
    const float* __restrict__ sig_y, const float* __restrict__ wts,
    float* __restrict__ out)
{
  // padded rows (+4 floats) to avoid LDS bank conflicts; ~51 KB total
  __shared__ __align__(16) float gy_s [NT_PER_WG][H_ + 4];
  __shared__ __align__(16) float gxs_s[NT_PER_WG][W_ + 4];
  __shared__ __align__(16) float x_s  [2][B_][KC + 4];
  __shared__ float sred[64];

  const int t    = threadIdx.x;
  const int lane = t & 31;
  const int wave = t >> 5;
  const int half = lane >> 4;       // 0: lanes 0-15, 1: lanes 16-31
  const int nl   = lane & 15;
  const int n0   = blockIdx.y * NT_PER_WG;

  // ---- per-WG separable mask tables: gy[n][h], gx[n][w] * s[n] ----
  {
    const int r = t & 31;
    const int n = n0 + r;
    float ssum = 0.0f;
    if (t < NT_PER_WG) {
      const float mu = mu_x[n];
      const float is = 1.0f / sig_x[n];
      for (int w = 0; w < W_; ++w) {
        float d = ((float)w * (1.0f / 127.0f) - mu) * is;
        float e = __expf(-0.5f * d * d);
        gxs_s[r][w] = e;
        ssum += e * e;
      }
    } else {
      const float mu = mu_y[n];
      const float is = 1.0f / sig_y[n];
      for (int h = 0; h < H_; ++h) {
        float d = ((float)h * (1.0f / 127.0f) - mu) * is;
        float e = __expf(-0.5f * d * d);
        gy_s[r][h] = e;
        ssum += e * e;
      }
    }
    sred[t] = ssum;
  }
  __syncthreads();
  if (t < NT_PER_WG) {
    // g / sqrt(C*Sy*Sx) * sqrt(C*H*W) = g * sqrt(H*W / (Sy*Sx))
    float scale = 128.0f * rsqrtf(sred[t] * sred[t + NT_PER_WG]);
    for (int w = 0; w < W_; ++w) gxs_s[t][w] *= scale;
  }

  const int nloc = wave * 16 + nl;          // 0..31 within WG
  const int ng   = n0 + nloc;               // global neuron
  const int ksub = 2 * half;                // operand K offset per half-wave

  v8f acc0 = {0.f, 0.f, 0.f, 0.f, 0.f, 0.f, 0.f, 0.f};   // batch rows 0..15
  v8f acc1 = {0.f, 0.f, 0.f, 0.f, 0.f, 0.f, 0.f, 0.f};   // batch rows 16..31

  const int ks_begin = blockIdx.x * KSLICE_LEN;
  // lane-private streaming pointer into weights: element (k+ksub, ng)
  const float* wp = wts + (size_t)(ks_begin + ksub) * N_ + ng;

  // stage x[0:32, kb:kb+KC] into LDS buffer `buf` (16B per request)
  auto prefetch = [&](int kb, int buf) {
    for (int i = t; i < B_ * (KC / 4); i += 64) {   // 512 float4 / chunk
      int row = i >> 4;                             // KC/4 = 16 float4 per row
      int c4  = (i & 15) << 2;
      ASYNC_COPY_B128(&x_s[buf][row][c4], x + (size_t)row * K_ + kb + c4);
    }
  };

  prefetch(ks_begin, 0);

  for (int c = 0; c < NCHUNK; ++c) {
    async_wait_all();      // chunk c resident (in-order ASYNCcnt, nothing newer)
    __syncthreads();       // all waves' data in; buffer (c+1)&1 free to overwrite
    if (c + 1 < NCHUNK) prefetch(ks_begin + (c + 1) * KC, (c + 1) & 1);

    const float (*xb)[KC + 4] = x_s[c & 1];
    const int kb = ks_begin + c * KC;

    #pragma unroll 4
    for (int kk = 0; kk < KC; kk += 4) {
      const int k  = kb + kk;
      const int hw = k & (H_ * W_ - 1);
      const int h  = hw >> 7;        // constant across the 4-k group
      const int w0 = hw & (W_ - 1);

      // B operand, lane holds (K = ksub + j, N = nl): weight * gy * gx
      float gyv = gy_s[nloc][h];
      v2f   gxv = *(const v2f*)&gxs_s[nloc][w0 + ksub];
      float wv0 = wp[0];
      float wv1 = wp[N_];
      wp += 4 * N_;

      v2f bfrag;
      bfrag.x = wv0 * gyv * gxv.x;
      bfrag.y = wv1 * gyv * gxv.y;

      // A operands, lane holds (M = nl [+16], K = ksub + j)
      v2f a0 = *(const v2f*)&xb[nl][kk + ksub];
      v2f a1 = *(const v2f*)&xb[nl + 16][kk + ksub];

      acc0 = __builtin_amdgcn_wmma_f32_16x16x4_f32(false, a0, false, bfrag,
                                                   (short)0, acc0, false, false);
      acc1 = __builtin_amdgcn_wmma_f32_16x16x4_f32(false, a1, false, bfrag,
                                                   (short)0, acc1, false, false);
    }
  }

  // C/D layout: VGPR r -> row M = r + 8*half, col N = nl; K-split -> atomic add
  #pragma unroll
  for (int r = 0; r < 8; ++r) {
    int b = r + 8 * half;
    atomicAdd(&out[b * N_ + ng],        acc0[r]);
    atomicAdd(&out[(b + 16) * N_ + ng], acc1[r]);
  }
}

extern "C" void kernel_launch(void* const* d_in, const int* in_sizes, int n_in,
                              void* d_out, int out_size, void* d_ws, size_t ws_size,
                              hipStream_t stream) {
  const float* x     = (const float*)d_in[0];
  const float* mu_x  = (const float*)d_in[1];
  const float* mu_y  = (const float*)d_in[2];
  const float* sig_x = (const float*)d_in[3];
  const float* sig_y = (const float*)d_in[4];
  const float* wts   = (const float*)d_in[5];
  float* out = (float*)d_out;

  zero_out_kernel<<<(out_size + 255) / 256, 256, 0, stream>>>(out, out_size);

  dim3 grid(KSLICES, N_ / NT_PER_WG);   // 16 k-slices x 32 neuron tiles
  adaptive_local2d_wmma_kernel<<<grid, 64, 0, stream>>>(
      x, mu_x, mu_y, sig_x, sig_y, wts, out);
}